// modeler_43336220016859
// MI455X (gfx1250) — compile-verified
//
#include <hip/hip_runtime.h>
#include <math.h>

#define NPN 60000
#define NAN_ 30000
#define NODE_TOT 90000
#define FTD 256
#define ATTD 128
#define KCHUNK 128
#define LTS 68  // padded K-stride for transposed attention LDS tile

typedef __attribute__((ext_vector_type(2))) float v2f;
typedef __attribute__((ext_vector_type(8))) float v8f;
typedef __attribute__((ext_vector_type(4))) unsigned int v4u;
typedef __attribute__((ext_vector_type(8))) int v8i;
typedef __attribute__((ext_vector_type(4))) int v4i;

// Tensor Data Mover availability (this toolchain: 6-arg builtin,
// (uint32x4 g0, int32x8 g1, int32x4 g2, int32x4 g3, int32x8 vaddr4, i32 cpol))
#if defined(__has_builtin)
#if __has_builtin(__builtin_amdgcn_tensor_load_to_lds) && \
    __has_builtin(__builtin_amdgcn_s_wait_tensorcnt)
#define CDNA5_HAS_TDM 1
#else
#define CDNA5_HAS_TDM 0
#endif
#else
#define CDNA5_HAS_TDM 0
#endif

// D = A(16x4 f32) * B(4x16 f32) + C(16x16 f32)   -- CDNA5 V_WMMA_F32_16X16X4_F32
__device__ __forceinline__ v8f wmma_f32x4(v2f a, v2f b, v8f c) {
  return __builtin_amdgcn_wmma_f32_16x16x4_f32(false, a, false, b, (short)0, c,
                                               false, false);
}

__device__ __forceinline__ float wave_reduce_add(float s) {
#pragma unroll
  for (int off = 16; off > 0; off >>= 1) s += __shfl_xor(s, off, 32);
  return s;
}

#if CDNA5_HAS_TDM
// DMA a [rows x 64] f32 tile (row stride 256 floats) from global into LDS,
// per CDNA5 ISA D# layout. Issued by the calling wave.
__device__ __forceinline__ void tdm_load_tile_64w(const float* gsrc,
                                                  float* ldsDst, int rows) {
  const unsigned long long ga = (unsigned long long)(uintptr_t)gsrc;
  const unsigned int ldsA = (unsigned int)(uintptr_t)(void*)ldsDst;
  v4u g0;
  g0.x = 1u;                                   // count=1, user mode, no gather
  g0.y = ldsA;                                 // lds_addr [63:32]
  g0.z = (unsigned int)ga;                     // global_addr [95:64]
  g0.w = ((unsigned int)(ga >> 32) & 0x01FFFFFFu) | (2u << 30);  // addr hi|type=2
  v8i g1;
  g1[0] = (int)(2u << 16);          // wg_mask=0, data_size=2 (4 bytes)
  g1[1] = (int)(256u << 16);        // tensor_dim0=256 (bits 79:48, low16)
  g1[2] = (int)(256u << 16);        // tensor_dim0 hi=0 | tensor_dim1=256 low16
  g1[3] = (int)(64u << 16);         // tensor_dim1 hi=0 | tile_dim0=64
  g1[4] = rows;                     // tile_dim1=rows, tile_dim2=0 (2D tile)
  g1[5] = 256;                      // tensor_dim0_stride low32 (floats)
  g1[6] = 0;                        // stride0 hi | stride1 low16 (65536 -> 0)
  g1[7] = 1;                        // stride1[47:16] = 65536>>16
  v4i gz4 = {0, 0, 0, 0};
  v8i gz8 = {0, 0, 0, 0, 0, 0, 0, 0};  // VADDR4: unused / NULL
  __builtin_amdgcn_tensor_load_to_lds(g0, g1, gz4, gz4, gz8, 0);
  __builtin_amdgcn_s_wait_tensorcnt(0);
}
#endif

// ---------------------------------------------------------------------------
// Segment scatter-add: agg[src] += feats[dst + dstOff], deg[src] += 1
// ---------------------------------------------------------------------------
__global__ __launch_bounds__(256) void scatter_kernel(
    const float* __restrict__ feats, const int* __restrict__ src,
    const int* __restrict__ dst, int nE, int dstOff, float* __restrict__ agg,
    float* __restrict__ deg) {
  int e = blockIdx.x * 8 + (threadIdx.x >> 5);
  if (e >= nE) return;
  int lane = threadIdx.x & 31;
  int s = src[e];
  size_t d = (size_t)(dst[e] + dstOff);
  const float* f = feats + d * FTD;
  float* o = agg + (size_t)s * FTD;
#pragma unroll
  for (int u = 0; u < 8; ++u) atomicAdd(o + lane * 8 + u, f[lane * 8 + u]);
  if (lane == 0) atomicAdd(deg + s, 1.0f);
}

__global__ __launch_bounds__(256) void normalize_kernel(
    float* __restrict__ agg, const float* __restrict__ deg, int rows) {
  int i = blockIdx.x * blockDim.x + threadIdx.x;  // rows*64 float4 lanes
  int n = i >> 6;
  if (n >= rows) return;
  float dinv = 1.0f / fmaxf(deg[n], 1.0f);
  float4* p = (float4*)(agg + (size_t)n * FTD) + (i & 63);
  float4 v = *p;
  v.x *= dinv; v.y *= dinv; v.z *= dinv; v.w *= dinv;
  *p = v;
}

// ---------------------------------------------------------------------------
// GEMM: C[M,256] = A1[M,256] @ W1[256,256] (+ A2 @ W2) + bias, epilogues:
//  EPI=0: PReLU store  EPI=1: plain store  EPI=2: sum((C-Ref)^2) -> accOut
// WG: 8 waves, 128 rows x 64 cols; wave tile 16x64; W chunk in 32KB LDS,
// filled by the Tensor Data Mover when available.
// ---------------------------------------------------------------------------
template <int EPI>
__global__ __launch_bounds__(256) void gemm_k256(
    const float* __restrict__ A1, const float* __restrict__ W1,
    const float* __restrict__ A2, const float* __restrict__ W2,
    const float* __restrict__ bias, const float* __restrict__ alphaPtr,
    float* __restrict__ C, const float* __restrict__ Ref,
    float* __restrict__ accOut, int M) {
  __shared__ float lw[KCHUNK * 64];
  const int lane = threadIdx.x & 31;
  const int wave = threadIdx.x >> 5;
  const int lo = lane & 15, hi = lane >> 4;
  const int rowBase = blockIdx.x * 128 + wave * 16;
  const int colBlk = blockIdx.y * 64;
  const int aRow = (rowBase + lo < M) ? (rowBase + lo) : (M - 1);

  v8f acc[4];
#pragma unroll
  for (int t = 0; t < 4; ++t)
    acc[t] = v8f{0.f, 0.f, 0.f, 0.f, 0.f, 0.f, 0.f, 0.f};

  const int nA = (A2 != nullptr) ? 2 : 1;
  for (int m = 0; m < nA; ++m) {
    const float* __restrict__ A = (m == 0) ? A1 : A2;
    const float* __restrict__ W = (m == 0) ? W1 : W2;
    const float* ap = A + (size_t)aRow * FTD + 2 * hi;
    for (int kc = 0; kc < FTD; kc += KCHUNK) {
      __syncthreads();  // previous chunk fully consumed
#if CDNA5_HAS_TDM
      if (threadIdx.x < 32)  // one TDM issue per workgroup (EXEC ignored)
        tdm_load_tile_64w(W + (size_t)kc * FTD + colBlk, lw, KCHUNK);
#else
      for (int idx = threadIdx.x; idx < KCHUNK * 16; idx += 256) {
        const int kk = idx >> 4, c4 = idx & 15;
        *(float4*)(lw + kk * 64 + c4 * 4) =
            *(const float4*)(W + (size_t)(kc + kk) * FTD + colBlk + c4 * 4);
      }
#endif
      __syncthreads();
      __builtin_prefetch(ap + kc + KCHUNK, 0, 1);  // global_prefetch_b8
#pragma unroll 2
      for (int ks = 0; ks < KCHUNK; ks += 4) {
        const float* app = ap + kc + ks;
        v2f a;
        a.x = app[0];
        a.y = app[1];
#pragma unroll
        for (int t = 0; t < 4; ++t) {
          v2f b;
          b.x = lw[(ks + 2 * hi) * 64 + t * 16 + lo];
          b.y = lw[(ks + 2 * hi + 1) * 64 + t * 16 + lo];
          acc[t] = wmma_f32x4(a, b, acc[t]);
        }
      }
    }
  }

  if (EPI == 2) {
    float s = 0.f;
#pragma unroll
    for (int t = 0; t < 4; ++t) {
      int col = colBlk + t * 16 + lo;
      float bv = bias ? bias[col] : 0.f;
#pragma unroll
      for (int v = 0; v < 8; ++v) {
        int row = rowBase + v + 8 * hi;
        if (row < M) {
          float d = acc[t][v] + bv - Ref[(size_t)row * FTD + col];
          s += d * d;
        }
      }
    }
    s = wave_reduce_add(s);
    if (lane == 0) atomicAdd(accOut, s);
  } else {
    float alpha = (EPI == 0 && alphaPtr) ? alphaPtr[0] : 0.f;
#pragma unroll
    for (int t = 0; t < 4; ++t) {
      int col = colBlk + t * 16 + lo;
      float bv = bias ? bias[col] : 0.f;
#pragma unroll
      for (int v = 0; v < 8; ++v) {
        int row = rowBase + v + 8 * hi;
        if (row < M) {
          float x = acc[t][v] + bv;
          if (EPI == 0) x = (x >= 0.f) ? x : alpha * x;
          C[(size_t)row * FTD + col] = x;
        }
      }
    }
  }
}

// ---------------------------------------------------------------------------
// Attention scores: e[row] = tanh(V[row,:] @ Ws + bs) . q   (N=128, K=256)
// WG: 4 waves, 64 rows; wave tile 16x128 (8 col tiles). Ws chunk kept
// K-transposed in LDS (pad 68) so each B fragment is one aligned ds_load_b64.
// ---------------------------------------------------------------------------
__global__ __launch_bounds__(128) void att_kernel(
    const float* __restrict__ V, const float* __restrict__ Ws,
    const float* __restrict__ bs, const float* __restrict__ q,
    float* __restrict__ e, int M) {
  __shared__ float lt[ATTD * LTS];  // [col][k] transposed, 34 KB
  const int lane = threadIdx.x & 31;
  const int wave = threadIdx.x >> 5;
  const int lo = lane & 15, hi = lane >> 4;
  const int rowBase = blockIdx.x * 64 + wave * 16;
  const int aRow = (rowBase + lo < M) ? (rowBase + lo) : (M - 1);

  v8f acc[8];
#pragma unroll
  for (int t = 0; t < 8; ++t)
    acc[t] = v8f{0.f, 0.f, 0.f, 0.f, 0.f, 0.f, 0.f, 0.f};

  const float* ap = V + (size_t)aRow * FTD + 2 * hi;
  for (int kc = 0; kc < FTD; kc += 64) {
    __syncthreads();
    for (int idx = threadIdx.x; idx < 64 * 32; idx += 128) {
      const int kk = idx >> 5, c4 = idx & 31;
      float4 v = *(const float4*)(Ws + (size_t)(kc + kk) * ATTD + c4 * 4);
      lt[(c4 * 4 + 0) * LTS + kk] = v.x;
      lt[(c4 * 4 + 1) * LTS + kk] = v.y;
      lt[(c4 * 4 + 2) * LTS + kk] = v.z;
      lt[(c4 * 4 + 3) * LTS + kk] = v.w;
    }
    __syncthreads();
    __builtin_prefetch(ap + kc + 64, 0, 1);
#pragma unroll 2
    for (int ks = 0; ks < 64; ks += 4) {
      const float* app = ap + kc + ks;
      v2f a;
      a.x = app[0];
      a.y = app[1];
#pragma unroll
      for (int t = 0; t < 8; ++t) {
        v2f b = *(const v2f*)(lt + (t * 16 + lo) * LTS + ks + 2 * hi);
        acc[t] = wmma_f32x4(a, b, acc[t]);
      }
    }
  }

  float part[8];
#pragma unroll
  for (int v = 0; v < 8; ++v) part[v] = 0.f;
#pragma unroll
  for (int t = 0; t < 8; ++t) {
    int col = t * 16 + lo;
    float bv = bs[col], qv = q[col];
#pragma unroll
    for (int v = 0; v < 8; ++v) part[v] += tanhf(acc[t][v] + bv) * qv;
  }
#pragma unroll
  for (int v = 0; v < 8; ++v) {
    float s = part[v];
#pragma unroll
    for (int off = 8; off > 0; off >>= 1) s += __shfl_xor(s, off, 16);
    int row = rowBase + v + 8 * hi;
    if (lo == 0 && row < M) e[row] = s;
  }
}

// softmax over 2 relations + weighted sum of vec rows -> summary (p-type)
__global__ __launch_bounds__(256) void summary_p_kernel(
    const float* __restrict__ vec, const float* __restrict__ e,
    float* __restrict__ summary, int N) {
  int i = blockIdx.x * blockDim.x + threadIdx.x;  // N*64 float4 lanes
  int n = i >> 6;
  if (n >= N) return;
  int c4 = i & 63;
  float e0 = e[n], e1 = e[N + n];
  float mx = fmaxf(e0, e1);
  float x0 = expf(e0 - mx), x1 = expf(e1 - mx);
  float inv = 1.0f / (x0 + x1);
  float b0 = x0 * inv, b1 = x1 * inv;
  float4 v0 = *(const float4*)(vec + (size_t)n * FTD + c4 * 4);
  float4 v1 = *(const float4*)(vec + (size_t)(N + n) * FTD + c4 * 4);
  float4 o;
  o.x = b0 * v0.x + b1 * v1.x;
  o.y = b0 * v0.y + b1 * v1.y;
  o.z = b0 * v0.z + b1 * v1.z;
  o.w = b0 * v0.w + b1 * v1.w;
  *(float4*)(summary + (size_t)n * FTD + c4 * 4) = o;
}

// pvec[i] = mean_j embs[neigh[i,j]]
__global__ __launch_bounds__(256) void pvec_kernel(
    const float* __restrict__ embs, const int* __restrict__ neigh,
    float* __restrict__ pvec, int B) {
  int i = blockIdx.x * 8 + (threadIdx.x >> 5);
  if (i >= B) return;
  int lane = threadIdx.x & 31;
  float s[8];
#pragma unroll
  for (int u = 0; u < 8; ++u) s[u] = 0.f;
  for (int j = 0; j < 16; ++j) {
    const float* f = embs + (size_t)neigh[i * 16 + j] * FTD + lane * 8;
#pragma unroll
    for (int u = 0; u < 8; ++u) s[u] += f[u];
  }
  float* o = pvec + (size_t)i * FTD + lane * 8;
#pragma unroll
  for (int u = 0; u < 8; ++u) o[u] = s[u] * (1.0f / 16.0f);
}

__global__ __launch_bounds__(256) void margin_kernel(
    const float* __restrict__ embs, const float* __restrict__ pvec,
    const int* __restrict__ batch, const int* __restrict__ shuf,
    float* __restrict__ acc, int B) {
  int i = blockIdx.x * 8 + (threadIdx.x >> 5);
  if (i >= B) return;
  int lane = threadIdx.x & 31;
  const float* v = embs + (size_t)batch[i] * FTD + lane * 8;
  const float* p1 = pvec + (size_t)i * FTD + lane * 8;
  const float* p2 = pvec + (size_t)shuf[i] * FTD + lane * 8;
  float s1 = 0.f, s2 = 0.f;
#pragma unroll
  for (int u = 0; u < 8; ++u) {
    float vv = v[u];
    s1 += vv * p1[u];
    s2 += vv * p2[u];
  }
  s1 = wave_reduce_add(s1);
  s2 = wave_reduce_add(s2);
  if (lane == 0) {
    float lp = 1.0f / (1.0f + expf(-s1));
    float ln = 1.0f / (1.0f + expf(-s2));
    float c = fmaxf(ln - lp + 0.5f, 0.f);
    atomicAdd(acc, c * (1.0f / (float)B));
  }
}

__global__ void finalize_kernel(const float* __restrict__ regAcc,
                                const float* __restrict__ margAcc,
                                float* __restrict__ out) {
  out[0] = 0.001f * regAcc[0] + margAcc[0];
}

// ---------------------------------------------------------------------------
extern "C" void kernel_launch(void* const* d_in, const int* in_sizes, int n_in,
                              void* d_out, int out_size, void* d_ws,
                              size_t ws_size, hipStream_t stream) {
  // --- input index mapping: insertion order vs JAX-sorted leaf order ---
  int iF, iW_p1, ib_p1, ia_p1, iWs_p1, ibs_p1, iq_p1, iWf_p1, ibf_p1;
  int iW_a1, ib_a1, ia_a1, iWs_a1, ibs_a1, iq_a1, iWf_a1, ibf_a1;
  int iWfc, ibfc, iPaS, iPaD, iPpS, iPpD, iApS, iApD, iBatch, iNeigh, iShuf;
  if (in_sizes[0] == NODE_TOT * FTD) {  // setup_inputs() insertion order
    iF = 0;
    iW_p1 = 9;  ib_p1 = 10; ia_p1 = 11; iWs_p1 = 12; ibs_p1 = 13; iq_p1 = 14;
    iWf_p1 = 15; ibf_p1 = 16;
    iW_a1 = 25; ib_a1 = 26; ia_a1 = 27; iWs_a1 = 28; ibs_a1 = 29; iq_a1 = 30;
    iWf_a1 = 31; ibf_a1 = 32;
    iWfc = 33; ibfc = 34;
    iPaS = 35; iPaD = 36; iPpS = 37; iPpD = 38; iApS = 39; iApD = 40;
    iBatch = 41; iNeigh = 42; iShuf = 43;
  } else {  // jax tree_leaves (sorted dict keys at every level)
    iApD = 0; iApS = 1; iBatch = 2; iF = 3; iNeigh = 4; iPaD = 5; iPaS = 6;
    // params: a.0 [7..14], a.1 [15..22], fc_p [23,24], p.0 [25..32], p.1 [33..40]
    iW_a1 = 15; iWf_a1 = 16; iWs_a1 = 17; ia_a1 = 18; ib_a1 = 19; ibf_a1 = 20;
    ibs_a1 = 21; iq_a1 = 22;
    iWfc = 23; ibfc = 24;
    iW_p1 = 33; iWf_p1 = 34; iWs_p1 = 35; ia_p1 = 36; ib_p1 = 37; ibf_p1 = 38;
    ibs_p1 = 39; iq_p1 = 40;
    iPpD = 41; iPpS = 42; iShuf = 43;
  }
  auto F = [&](int i) { return (const float*)d_in[i]; };
  auto I = [&](int i) { return (const int*)d_in[i]; };

  const float* feats = F(iF);
  const int Epa = in_sizes[iPaS], Epp = in_sizes[iPpS], Eap = in_sizes[iApS];
  const int B = in_sizes[iBatch];

  // --- workspace layout (floats) ---
  float* ws = (float*)d_ws;
  const size_t RELROWS = (size_t)2 * NPN + NAN_;  // 150000
  float* agg = ws;                                // 150000*256
  float* vec = agg + RELROWS * FTD;               // 150000*256
  float* evec = vec + RELROWS * FTD;              // 120000
  float* deg = evec + (size_t)2 * NPN;            // 150000
  float* pvec = deg + RELROWS;                    // 4096*256
  float* regAcc = pvec + (size_t)B * FTD;
  float* margAcc = regAcc + 1;

  float* embs = (float*)d_out;
  float* totalOut = embs + (size_t)NODE_TOT * FTD;

  (void)hipMemsetAsync(agg, 0, RELROWS * FTD * sizeof(float), stream);
  (void)hipMemsetAsync(deg, 0, RELROWS * sizeof(float), stream);
  (void)hipMemsetAsync(regAcc, 0, 2 * sizeof(float), stream);

  // --- segment means (3 relations; layer-0 params are dead code, skipped) ---
  scatter_kernel<<<(Epa + 7) / 8, 256, 0, stream>>>(feats, I(iPaS), I(iPaD),
                                                    Epa, NPN, agg, deg);
  scatter_kernel<<<(Epp + 7) / 8, 256, 0, stream>>>(
      feats, I(iPpS), I(iPpD), Epp, 0, agg + (size_t)NPN * FTD, deg + NPN);
  scatter_kernel<<<(Eap + 7) / 8, 256, 0, stream>>>(
      feats, I(iApS), I(iApD), Eap, 0, agg + (size_t)2 * NPN * FTD,
      deg + 2 * NPN);
  normalize_kernel<<<(unsigned)((RELROWS * 64 + 255) / 256), 256, 0, stream>>>(
      agg, deg, (int)RELROWS);

  // --- vec = prelu(agg @ W + b) ---
  {
    dim3 g((2 * NPN + 127) / 128, 4);
    gemm_k256<0><<<g, 256, 0, stream>>>(agg, F(iW_p1), nullptr, nullptr,
                                        F(ib_p1), F(ia_p1), vec, nullptr,
                                        nullptr, 2 * NPN);
  }
  {
    dim3 g((NAN_ + 127) / 128, 4);
    gemm_k256<0><<<g, 256, 0, stream>>>(
        agg + (size_t)2 * NPN * FTD, F(iW_a1), nullptr, nullptr, F(ib_a1),
        F(ia_a1), vec + (size_t)2 * NPN * FTD, nullptr, nullptr, NAN_);
  }

  // --- semantic attention (p only; 'a' has R=1 => summary == vec_a) ---
  att_kernel<<<(2 * NPN + 63) / 64, 128, 0, stream>>>(vec, F(iWs_p1), F(ibs_p1),
                                                      F(iq_p1), evec, 2 * NPN);
  float* summary_p = agg;  // reuse agg slot 0 (no longer needed)
  summary_p_kernel<<<(NPN * 64 + 255) / 256, 256, 0, stream>>>(vec, evec,
                                                               summary_p, NPN);

  // --- embs = concat(summary, feats) @ Wf + bf   (split-K dual GEMM) ---
  {
    dim3 g((NPN + 127) / 128, 4);
    gemm_k256<1><<<g, 256, 0, stream>>>(
        summary_p, F(iWf_p1), feats, F(iWf_p1) + (size_t)FTD * FTD, F(ibf_p1),
        nullptr, embs, nullptr, nullptr, NPN);
  }
  {
    dim3 g((NAN_ + 127) / 128, 4);
    gemm_k256<1><<<g, 256, 0, stream>>>(
        vec + (size_t)2 * NPN * FTD, F(iWf_a1), feats + (size_t)NPN * FTD,
        F(iWf_a1) + (size_t)FTD * FTD, F(ibf_a1), nullptr,
        embs + (size_t)NPN * FTD, nullptr, nullptr, NAN_);
  }

  // --- reg loss: sum((embs @ fcW + fcb - feats)^2) over all nodes ---
  {
    dim3 g((NODE_TOT + 127) / 128, 4);
    gemm_k256<2><<<g, 256, 0, stream>>>(embs, F(iWfc), nullptr, nullptr,
                                        F(ibfc), nullptr, nullptr, feats,
                                        regAcc, NODE_TOT);
  }

  // --- margin loss ---
  pvec_kernel<<<(B + 7) / 8, 256, 0, stream>>>(embs, I(iNeigh), pvec, B);
  margin_kernel<<<(B + 7) / 8, 256, 0, stream>>>(embs, pvec, I(iBatch),
                                                 I(iShuf), margAcc, B);
  finalize_kernel<<<1, 1, 0, stream>>>(regAcc, margAcc, totalOut);
}